// Matcher_80839874445716
// MI455X (gfx1250) — compile-verified
//
#include <hip/hip_runtime.h>
#include <cstdint>

// ---------------------------------------------------------------------------
// Zero-FLOP broadcast/scatter kernel. Store-bandwidth bound (~350 MB out vs
// ~3 MB in): optimize for wide non-temporal global_store_b128 streams.
// CDNA5 paths used: NT b128 stores, global_prefetch, and
// GLOBAL_LOAD_ASYNC_TO_LDS_B128 + s_wait_asynccnt staging.
// ---------------------------------------------------------------------------

typedef float v4f __attribute__((ext_vector_type(4)));

#define K_FEAT   64          // reference K (fixed)
#define K4       16          // float4 chunks per row
#define NTHREADS 256
#define SMEM_CHUNKS 2048     // 32 KB LDS staging (covers M2 <= 128)

#if defined(__AMDGCN__) && \
    __has_builtin(__builtin_amdgcn_global_load_async_to_lds_b128) && \
    __has_builtin(__builtin_amdgcn_s_wait_asynccnt)
#define USE_ASYNC_LDS 1
// Match the builtin's parameter types exactly (per hipcc diagnostic):
//   param0: global (AS1) pointer to int __attribute__((vector_size(16))), non-const
//   param1: LDS    (AS3) pointer to the same vector type
typedef int v4i_gcc __attribute__((vector_size(16)));
typedef __attribute__((address_space(1))) v4i_gcc gas_v4i;
typedef __attribute__((address_space(3))) v4i_gcc las_v4i;
#endif

// ---------------------------------------------------------------------------
// Setup: per-graph counts & exclusive starts for ids1 / ids2.
// meta layout (ints): [0,G) counts1 | [G,2G) starts1 | [2G,3G) counts2 | [3G,4G) starts2
// Single block; G <= 256 assumed (reference: G = 32).
// ---------------------------------------------------------------------------
__global__ __launch_bounds__(NTHREADS) void setup_kernel(
    const int* __restrict__ ids1, int n1,
    const int* __restrict__ ids2, int n2,
    const int* __restrict__ dG,
    int* __restrict__ meta)
{
    __shared__ int cnt[512];                 // [0,256): ids1, [256,512): ids2
    const int tid = threadIdx.x;
    for (int idx = tid; idx < 512; idx += NTHREADS) cnt[idx] = 0;
    __syncthreads();
    for (int n = tid; n < n1; n += NTHREADS) atomicAdd(&cnt[ids1[n] & 255], 1);
    for (int n = tid; n < n2; n += NTHREADS) atomicAdd(&cnt[256 + (ids2[n] & 255)], 1);
    __syncthreads();
    if (tid == 0) {
        const int G = *dG;
        int run = 0;
        for (int g = 0; g < G; ++g) { meta[g] = cnt[g]; meta[G + g] = run; run += cnt[g]; }
        run = 0;
        for (int g = 0; g < G; ++g) { meta[2*G + g] = cnt[256 + g]; meta[3*G + g] = run; run += cnt[256 + g]; }
    }
}

// ---------------------------------------------------------------------------
// Broadcast kernel. Output layout per tensor t in {f,e,v}: [G, 2, M1, M2, K].
// Strip s = ((t*G + g)*2 + ch)*M1 + i  ->  M2*K contiguous floats.
//   ch==0: out[...] = row1 = x1[start1[g]+i, :] repeated M2 times (zeros if padded row)
//   ch==1: out[...] = slab2 = x2[start2[g] .. +count2[g], :] zero-padded to M2 rows
// ---------------------------------------------------------------------------
__global__ __launch_bounds__(NTHREADS) void bcast_kernel(
    const float* __restrict__ xf1, const float* __restrict__ xe1, const float* __restrict__ xv1,
    const float* __restrict__ xf2, const float* __restrict__ xe2, const float* __restrict__ xv2,
    const int* __restrict__ meta,
    const int* __restrict__ dG, const int* __restrict__ dM1, const int* __restrict__ dM2,
    float* __restrict__ out)
{
    const int G  = *dG;
    const int M1 = *dM1;
    const int M2 = *dM2;
    const int nStrips = 3 * G * 2 * M1;
    const int chunks  = M2 * K4;          // float4 chunks per strip
    const int tid = threadIdx.x;
    v4f* __restrict__ out4 = (v4f*)out;
    const v4f vzero = {0.f, 0.f, 0.f, 0.f};

#ifdef USE_ASYNC_LDS
    __shared__ v4f smem[SMEM_CHUNKS];
#endif

    for (int s = blockIdx.x; s < nStrips; s += gridDim.x) {
        int i = s % M1;
        int r = s / M1;
        const int ch = r & 1; r >>= 1;
        const int g = r % G;
        const int t = r / G;

        v4f* __restrict__ dst = out4 + (size_t)s * (size_t)chunks;
        const int* counts = meta + (ch ? 2 * G : 0);
        const int* starts = counts + G;

        if (ch == 0) {
            const float* x1 = (t == 0) ? xf1 : (t == 1) ? xe1 : xv1;
            const v4f* x1v = (const v4f*)x1;
            v4f v = vzero;
            if (i < counts[g])                       // padded rows stay zero
                v = x1v[(size_t)(starts[g] + i) * K4 + (tid & (K4 - 1))];
            // stride NTHREADS is a multiple of K4, so (c % K4) is per-thread const
            for (int c = tid; c < chunks; c += NTHREADS)
                __builtin_nontemporal_store(v, dst + c);
        } else {
            const float* x2 = (t == 0) ? xf2 : (t == 1) ? xe2 : xv2;
            const v4f* slab = (const v4f*)x2 + (size_t)starts[g] * K4;
            const int valid = counts[g] * K4;        // in-bounds float4 chunks
            __builtin_prefetch(slab, 0, 1);
#ifdef USE_ASYNC_LDS
            if (chunks <= SMEM_CHUNKS) {
                __syncthreads();                     // smem reuse across strips
                for (int c = tid; c < chunks; c += NTHREADS) {
                    if (c < valid) {
                        __builtin_amdgcn_global_load_async_to_lds_b128(
                            (gas_v4i*)(uintptr_t)(slab + c),
                            (las_v4i*)(uint32_t)(uintptr_t)(smem + c),
                            0, 0);
                    } else {
                        smem[c] = vzero;             // zero padding
                    }
                }
                __builtin_amdgcn_s_wait_asynccnt(0);
                __syncthreads();
                for (int c = tid; c < chunks; c += NTHREADS)
                    __builtin_nontemporal_store(smem[c], dst + c);
            } else
#endif
            {
                for (int c = tid; c < chunks; c += NTHREADS) {
                    v4f v = (c < valid) ? slab[c] : vzero;
                    __builtin_nontemporal_store(v, dst + c);
                }
            }
        }
    }
}

// ---------------------------------------------------------------------------
// d_in order (setup_inputs dict): x_f_1, x_e_1, x_v_1, x_f_2, x_e_2, x_v_2,
//                                 ids1, ids2, n_graphs, maxcount1, maxcount2
// ---------------------------------------------------------------------------
extern "C" void kernel_launch(void* const* d_in, const int* in_sizes, int n_in,
                              void* d_out, int out_size, void* d_ws, size_t ws_size,
                              hipStream_t stream) {
    const float* xf1 = (const float*)d_in[0];
    const float* xe1 = (const float*)d_in[1];
    const float* xv1 = (const float*)d_in[2];
    const float* xf2 = (const float*)d_in[3];
    const float* xe2 = (const float*)d_in[4];
    const float* xv2 = (const float*)d_in[5];
    const int* ids1 = (const int*)d_in[6];
    const int* ids2 = (const int*)d_in[7];
    const int* dG   = (const int*)d_in[8];
    const int* dM1  = (const int*)d_in[9];
    const int* dM2  = (const int*)d_in[10];
    const int n1 = in_sizes[6];
    const int n2 = in_sizes[7];

    int* meta = (int*)d_ws;   // 4*G ints

    setup_kernel<<<1, NTHREADS, 0, stream>>>(ids1, n1, ids2, n2, dG, meta);

    // Dims live on-device; fixed grid + stride loop keeps capture deterministic.
    // Typical nStrips = 3*32*2*~84 ~= 16k; each strip ~= M2*256B of NT stores.
    bcast_kernel<<<16384, NTHREADS, 0, stream>>>(
        xf1, xe1, xv1, xf2, xe2, xv2, meta, dG, dM1, dM2, (float*)d_out);
}